// VolumeRender_10608569221409
// MI455X (gfx1250) — compile-verified
//
#include <hip/hip_runtime.h>
#include <hip/hip_bf16.h>

#ifndef __has_builtin
#define __has_builtin(x) 0
#endif

#if __has_builtin(__builtin_amdgcn_global_load_async_to_lds_b128)
#define HAVE_ASYNC_LDS 1
#else
#define HAVE_ASYNC_LDS 0
#endif

#if HAVE_ASYNC_LDS
#if __has_builtin(__builtin_amdgcn_s_wait_asynccnt)
#define WAIT_ASYNC(n) do { __builtin_amdgcn_s_wait_asynccnt(n); asm volatile("" ::: "memory"); } while (0)
#else
#define WAIT_ASYNC(n) asm volatile("s_wait_asynccnt %0" :: "n"(n) : "memory")
#endif
#else
#define WAIT_ASYNC(n) ((void)0)
#endif

// The async-to-LDS builtin takes int4-vector pointers in AS1 (global) / AS3 (LDS).
typedef int v4i __attribute__((ext_vector_type(4)));
typedef __attribute__((address_space(1))) v4i gv4i_t;
typedef __attribute__((address_space(3))) v4i lv4i_t;

// Global pointers: flat VA == global VA on AMDGPU, so inttoptr into AS1 is safe.
#define TO_GLOBAL(p) ((gv4i_t*)(uintptr_t)(p))
// LDS: genuine addrspacecast from the generic pointer to a __shared__ object.
#define TO_LDS(p)    ((lv4i_t*)(p))

// One wave32 per ray. Chunks of 32 samples staged through LDS (double buffered).
// Per-chunk LDS layout (bytes): [0,128) sigma  [128,256) delta  [256,384) ts
//                               [384,768) rgb (32 samples * 12B)
__global__ __launch_bounds__(256) void volrend_fwd_kernel(
    const float* __restrict__ sigmas, const float* __restrict__ rgbs,
    const float* __restrict__ deltas, const float* __restrict__ ts,
    const int*   __restrict__ rays_a, const float* __restrict__ thr_p,
    int*   __restrict__ out_ns,   float* __restrict__ out_opac,
    float* __restrict__ out_depth, float* __restrict__ out_rgb,
    float* __restrict__ out_w,    int n_rays)
{
    __shared__ __align__(16) float lbuf[8 * 2 * 192];  // 8 waves * 2 buffers * 768B

    const int lane = threadIdx.x & 31;
    const int wid  = threadIdx.x >> 5;
    const int r    = blockIdx.x * 8 + wid;
    if (r >= n_rays) return;

    const int   start = rays_a[r * 3 + 1];
    const int   count = rays_a[r * 3 + 2];
    const float thr   = thr_p[0];
    const int   nch   = count >> 5;            // counts are multiples of 32

    float* wbase = &lbuf[wid * 384];

    auto stage = [&](int chunk, float* buf) {
        const long long smp  = (long long)start + (long long)chunk * 32;
        const char* sigb = (const char*)sigmas + smp * 4;
        const char* delb = (const char*)deltas + smp * 4;
        const char* tsb  = (const char*)ts     + smp * 4;
        const char* rgbb = (const char*)rgbs   + smp * 12;
        const int g = lane >> 3, l8 = lane & 7;
        const char* gp; int loff;
        if      (g == 0) { gp = sigb + l8 * 16; loff =   0 + l8 * 16; }
        else if (g == 1) { gp = delb + l8 * 16; loff = 128 + l8 * 16; }
        else if (g == 2) { gp = tsb  + l8 * 16; loff = 256 + l8 * 16; }
        else             { gp = rgbb + l8 * 16; loff = 384 + l8 * 16; }
#if HAVE_ASYNC_LDS
        __builtin_amdgcn_global_load_async_to_lds_b128(
            TO_GLOBAL(gp), TO_LDS((char*)buf + loff), 0, 0);
        if (lane < 16)
            __builtin_amdgcn_global_load_async_to_lds_b128(
                TO_GLOBAL(rgbb + 128 + lane * 16),
                TO_LDS((char*)buf + 512 + lane * 16), 0, 0);
#else
        *(float4*)((char*)buf + loff) = *(const float4*)gp;
        if (lane < 16)
            *(float4*)((char*)buf + 512 + lane * 16) =
                *(const float4*)(rgbb + 128 + lane * 16);
#endif
    };

    float cum = 0.f, op = 0.f, dep = 0.f, cr = 0.f, cg = 0.f, cb = 0.f;
    int ns = 0;

    if (nch > 0) stage(0, wbase);

    for (int i = 0; i < nch; ++i) {
        float* cur = wbase + (i & 1) * 192;
        if (i + 1 < nch) { stage(i + 1, wbase + ((i + 1) & 1) * 192); WAIT_ASYNC(2); }
        else             { WAIT_ASYNC(0); }

        const float sg = cur[lane];
        const float dl = cur[32 + lane];
        const float tt = cur[64 + lane];
        const float c0 = cur[96 + lane * 3 + 0];
        const float c1 = cur[96 + lane * 3 + 1];
        const float c2 = cur[96 + lane * 3 + 2];

        const float sd = sg * dl;

        // wave32 inclusive scan of sd (5 shuffle steps)
        float incl = sd;
        #pragma unroll
        for (int d = 1; d < 32; d <<= 1) {
            float o = __shfl_up(incl, d, 32);
            if (lane >= d) incl += o;
        }

        const float T     = __expf(-(cum + incl - sd));  // exclusive prefix + carry
        const float a     = 1.0f - __expf(-sd);
        const bool  alive = (T >= thr);
        const float w     = alive ? a * T : 0.0f;

        const long long gi = (long long)start + (long long)i * 32 + lane;
        __builtin_nontemporal_store(w, out_w + gi);      // write-once stream: NT

        ns  += alive ? 1 : 0;
        op  += w;
        dep += w * tt;
        cr  += w * c0;
        cg  += w * c1;
        cb  += w * c2;

        cum += __shfl(incl, 31, 32);                     // carry chunk total
    }

    // wave32 reductions of the six per-ray accumulators
    #pragma unroll
    for (int d = 16; d >= 1; d >>= 1) {
        ns  += __shfl_down(ns,  d, 32);
        op  += __shfl_down(op,  d, 32);
        dep += __shfl_down(dep, d, 32);
        cr  += __shfl_down(cr,  d, 32);
        cg  += __shfl_down(cg,  d, 32);
        cb  += __shfl_down(cb,  d, 32);
    }

    if (lane == 0) {
        out_ns[r]          = ns;     // int32 bit pattern, matches reference dtype
        out_opac[r]        = op;
        out_depth[r]       = dep;
        out_rgb[3 * r + 0] = cr;
        out_rgb[3 * r + 1] = cg;
        out_rgb[3 * r + 2] = cb;
    }
}

extern "C" void kernel_launch(void* const* d_in, const int* in_sizes, int n_in,
                              void* d_out, int out_size, void* d_ws, size_t ws_size,
                              hipStream_t stream) {
    const float* sigmas = (const float*)d_in[0];
    const float* rgbs   = (const float*)d_in[1];
    const float* deltas = (const float*)d_in[2];
    const float* ts     = (const float*)d_in[3];
    const int*   rays_a = (const int*)  d_in[4];
    const float* thr    = (const float*)d_in[5];

    const int n_rays = in_sizes[4] / 3;

    float* out     = (float*)d_out;
    int*   out_ns  = (int*)d_out;                 // total_samples (int32)
    float* out_op  = out + (size_t)n_rays;        // opacity
    float* out_dep = out + (size_t)2 * n_rays;    // depth
    float* out_rgb = out + (size_t)3 * n_rays;    // rgb, [r][3]
    float* out_w   = out + (size_t)6 * n_rays;    // per-sample weights

    const int blocks = (n_rays + 7) / 8;          // 8 waves (rays) per 256-thread block
    volrend_fwd_kernel<<<blocks, 256, 0, stream>>>(
        sigmas, rgbs, deltas, ts, rays_a, thr,
        out_ns, out_op, out_dep, out_rgb, out_w, n_rays);
}